// PointNetSetAbstraction_446676598905
// MI455X (gfx1250) — compile-verified
//
#include <hip/hip_runtime.h>

// ---------------------------------------------------------------------------
// PointNet set-abstraction (temporal) for MI455X / gfx1250.
//  - query_ball: 1 thread per (branch, b, s), stable top-32 insertion.
//  - MLP layers: fp32 WMMA (v_wmma_f32_16x16x4_f32). One wave computes a
//    16-row x Cout strip: A fragment loaded once per K-step, reused across
//    Cout/16 accumulators -> X streamed from HBM exactly once per layer.
//  - Weights pre-padded to K multiple of 4 (no guards in the hot loop).
//  - BatchNorm: deterministic 2-stage reduction (no atomics), fused norm+relu.
//  - Max-pool over ns + transposed writes to d_out.
// ---------------------------------------------------------------------------

typedef __attribute__((ext_vector_type(2))) float v2f;
typedef __attribute__((ext_vector_type(8))) float v8f;

#define B_    4
#define N_    8192
#define S_    1024
#define D_    64
#define T_    16
#define NS_   32
#define M_    (B_ * S_ * NS_)   // 131072 rows
#define K0_   67
#define K0P_  68                 // padded to multiple of 4
#define EPS_  1e-5f
#define BIGV  8192.0f            // == float(N), sentinel used by reference
#define RADIUS2 0.002197265625f  // (12/256)^2
#define NCHUNK 256               // BN partial-reduction chunks (M_/NCHUNK=512)

// ---------------------------------------------------------------- query ball
__global__ void query_ball_kernel(const float* __restrict__ xyz,
                                  const float* __restrict__ center,
                                  const float* __restrict__ tlist,
                                  int* __restrict__ idx_out) {
  int tid = blockIdx.x * blockDim.x + threadIdx.x;
  if (tid >= 2 * B_ * S_) return;
  int br  = tid / (B_ * S_);
  int rem = tid - br * B_ * S_;
  int b   = rem / S_;
  float w0 = (br == 0) ? 1.0f : 0.5f;
  float w1 = (br == 0) ? 0.0f : 0.5f;

  const float* cen = center + (size_t)rem * 3;
  float cx = cen[0], cy = cen[1], ct = cen[2];
  float cxy2 = cx * cx + cy * cy;           // mimic a^2+b^2-2ab form of reference
  float ct2  = ct * ct;

  // max_t = (max consecutive diff of t_list[b])^2
  const float* tl = tlist + b * T_;
  float md = tl[1] - tl[0];
  for (int i = 1; i < T_ - 1; ++i) md = fmaxf(md, tl[i + 1] - tl[i]);
  float max_t = md * md;

  float dbest[NS_];
  int   ibest[NS_];
#pragma unroll
  for (int k = 0; k < NS_; ++k) { dbest[k] = BIGV; ibest[k] = 0; }

  float bestOri = 3.4e38f;
  int   firstIdx = 0;

  const float* xb = xyz + (size_t)b * N_ * 3;
  for (int n = 0; n < N_; ++n) {
    float x = xb[n * 3 + 0];
    float y = xb[n * 3 + 1];
    float t = xb[n * 3 + 2];
    float dxy = cxy2 + (x * x + y * y) - 2.0f * (cx * x + cy * y);
    float dtt = ct2 + t * t - 2.0f * (ct * t);
    float dori = (cxy2 + ct2) + (x * x + y * y + t * t)
               - 2.0f * (cx * x + cy * y + ct * t);
    if (dori < bestOri) { bestOri = dori; firstIdx = n; }
    if (dxy <= RADIUS2 && dtt <= max_t) {
      float dist = dxy * w0 + dtt * w1;
      if (dist < dbest[NS_ - 1]) {           // stable: equal keys keep old first
        int p = NS_ - 1;
        while (p > 0 && dbest[p - 1] > dist) {
          dbest[p] = dbest[p - 1];
          ibest[p] = ibest[p - 1];
          --p;
        }
        dbest[p] = dist;
        ibest[p] = n;
      }
    }
  }

  int* out = idx_out + (size_t)tid * NS_;    // layout [2][B][S][NS]
#pragma unroll
  for (int k = 0; k < NS_; ++k)
    out[k] = (dbest[k] >= BIGV) ? firstIdx : ibest[k];
}

// ------------------------------------------------------------------- gather
// Build X0 [M, 68]: 3 relative coords + 64 point feats + 1 zero pad column.
__global__ void gather_kernel(const float* __restrict__ xyz,
                              const float* __restrict__ points,
                              const float* __restrict__ center,
                              const int* __restrict__ idx,
                              float* __restrict__ X) {
  size_t t = (size_t)blockIdx.x * blockDim.x + threadIdx.x;
  if (t >= (size_t)M_ * K0P_) return;
  int    c  = (int)(t % K0P_);
  size_t r  = t / K0P_;                  // (b*S+s)*NS + j
  int    bs = (int)(r / NS_);            // b*S + s
  int    b  = bs / S_;
  int    n  = idx[r];
  float  v;
  if (c < 3)
    v = xyz[((size_t)b * N_ + n) * 3 + c] - center[(size_t)bs * 3 + c];
  else if (c < 3 + D_)
    v = points[((size_t)b * N_ + n) * D_ + (c - 3)];
  else
    v = 0.0f;
  X[r * K0P_ + c] = v;
}

// --------------------------------------------------------- weight pre-padding
// Wpad[Cout, Kp] = W[Cout, Kw] zero-extended in K -> no guards in GEMM loop.
__global__ void pad_weights_kernel(const float* __restrict__ W,
                                   float* __restrict__ Wpad,
                                   int Kw, int Kp, int Cout) {
  int t = blockIdx.x * blockDim.x + threadIdx.x;
  if (t >= Cout * Kp) return;
  int o = t / Kp;
  int k = t - o * Kp;
  Wpad[t] = (k < Kw) ? W[(size_t)o * Kw + k] : 0.0f;
}

// ------------------------------------------------------------- WMMA GEMM f32
// Y[M, COUT] = X[M, KP] * Wpad[COUT, KP]^T (+bias via C-init).
// One wave -> one 16-row strip across all COUT columns. A frag shared
// across NT = COUT/16 accumulators, so X is read exactly once.
template <int KW, int COUT>
__global__ void gemm_wmma_kernel(const float* __restrict__ X,
                                 const float* __restrict__ Wp,
                                 const float* __restrict__ bias,
                                 float* __restrict__ Y) {
  constexpr int KP = (KW + 3) & ~3;
  constexpr int NT = COUT / 16;

  int wave = blockIdx.x * (blockDim.x >> 5) + (threadIdx.x >> 5);  // == m-tile
  int lane = threadIdx.x & 31;
  int half = lane >> 4;                  // K-half selector (0 -> K{0,1}, 1 -> K{2,3})
  int l15  = lane & 15;
  int rowA = (wave << 4) + l15;          // A: M row held by this lane

  const v2f* xrow = reinterpret_cast<const v2f*>(
      X + (size_t)rowA * KP + (half << 1));
  const v2f* wrow[NT];
#pragma unroll
  for (int t = 0; t < NT; ++t)
    wrow[t] = reinterpret_cast<const v2f*>(
        Wp + (size_t)(t * 16 + l15) * KP + (half << 1));

  v8f acc[NT];
#pragma unroll
  for (int t = 0; t < NT; ++t) {
    float bb = bias[t * 16 + l15];       // fold bias into C fragment
    acc[t] = (v8f){bb, bb, bb, bb, bb, bb, bb, bb};
  }

#pragma unroll 4
  for (int k = 0; k < KP; k += 4) {
    v2f a = xrow[k >> 1];                // one A load per K-step, reused NT times
#pragma unroll
    for (int t = 0; t < NT; ++t) {
      v2f bv = wrow[t][k >> 1];
      acc[t] = __builtin_amdgcn_wmma_f32_16x16x4_f32(
          false, a, false, bv, (short)0, acc[t], false, false);
    }
  }

#pragma unroll
  for (int t = 0; t < NT; ++t) {
#pragma unroll
    for (int r = 0; r < 8; ++r) {        // C/D layout: vgpr r -> M=r (+8 hi lanes)
      int m = (wave << 4) + (half << 3) + r;
      Y[(size_t)m * COUT + t * 16 + l15] = acc[t][r];
    }
  }
}

// ------------------------------------------------------- BN (deterministic)
__global__ void bn_partial_kernel(const float* __restrict__ Y,
                                  float* __restrict__ partial, int C) {
  int c = threadIdx.x;                   // blockDim.x == C (coalesced reads)
  int chunk = blockIdx.x;                // NCHUNK chunks
  const int rows = M_ / NCHUNK;
  const float* p = Y + (size_t)chunk * rows * C + c;
  float s = 0.0f, s2 = 0.0f;
  for (int r = 0; r < rows; ++r) {
    float v = p[(size_t)r * C];
    s += v;
    s2 += v * v;
  }
  partial[((size_t)chunk * C + c) * 2 + 0] = s;
  partial[((size_t)chunk * C + c) * 2 + 1] = s2;
}

__global__ void bn_final_kernel(const float* __restrict__ partial,
                                float* __restrict__ meanvar, int C) {
  int c = threadIdx.x;                   // blockDim.x == C, grid == 1
  float s = 0.0f, s2 = 0.0f;
  for (int k = 0; k < NCHUNK; ++k) {     // fixed order -> deterministic
    s  += partial[((size_t)k * C + c) * 2 + 0];
    s2 += partial[((size_t)k * C + c) * 2 + 1];
  }
  float mean = s / (float)M_;
  float var  = s2 / (float)M_ - mean * mean;  // biased, as jnp.var
  meanvar[2 * c + 0] = mean;
  meanvar[2 * c + 1] = rsqrtf(var + EPS_);
}

__global__ void bn_apply_kernel(float* __restrict__ Y,
                                const float* __restrict__ meanvar,
                                const float* __restrict__ gamma,
                                const float* __restrict__ beta, int C) {
  size_t t = (size_t)blockIdx.x * blockDim.x + threadIdx.x;
  if (t >= (size_t)M_ * C) return;
  int c = (int)(t % C);
  float v = Y[t];
  v = gamma[c] * (v - meanvar[2 * c]) * meanvar[2 * c + 1] + beta[c];
  Y[t] = fmaxf(v, 0.0f);
}

// --------------------------------------------------------------- epilogues
__global__ void maxpool_out_kernel(const float* __restrict__ Y,
                                   float* __restrict__ out, int br) {
  int t = blockIdx.x * blockDim.x + threadIdx.x;
  if (t >= B_ * S_ * 128) return;
  int c  = t % 128;
  int bs = t / 128;
  int b  = bs / S_;
  int s  = bs - b * S_;
  const float* p = Y + (size_t)bs * NS_ * 128 + c;
  float m = p[0];
  for (int j = 1; j < NS_; ++j) m = fmaxf(m, p[(size_t)j * 128]);
  // res_points [B,2,128,S] lives after res_xyz [B,2,3,S] in d_out
  out[(size_t)B_ * 2 * 3 * S_ + ((size_t)(b * 2 + br) * 128 + c) * S_ + s] = m;
}

__global__ void center_out_kernel(const float* __restrict__ center,
                                  float* __restrict__ out) {
  int t = blockIdx.x * blockDim.x + threadIdx.x;
  if (t >= 2 * B_ * S_ * 3) return;
  int k  = t % 3;
  int r2 = t / 3;
  int s  = r2 % S_;
  int r3 = r2 / S_;
  int b  = r3 % B_;
  int br = r3 / B_;
  out[((size_t)(b * 2 + br) * 3 + k) * S_ + s] =
      center[((size_t)b * S_ + s) * 3 + k];
}

// ------------------------------------------------------------------ driver
template <int KW, int COUT>
static void run_layer(const float* X, const float* W, const float* bia,
                      const float* g, const float* be, float* Y,
                      float* Wpad, float* partial, float* meanvar,
                      hipStream_t stream) {
  constexpr int KP = (KW + 3) & ~3;
  pad_weights_kernel<<<(COUT * KP + 255) / 256, 256, 0, stream>>>(
      W, Wpad, KW, KP, COUT);
  // M_/16 = 8192 waves, 8 waves (256 threads) per block -> 1024 blocks, exact.
  gemm_wmma_kernel<KW, COUT><<<(M_ / 16) / 8, 256, 0, stream>>>(X, Wpad, bia, Y);
  bn_partial_kernel<<<NCHUNK, COUT, 0, stream>>>(Y, partial, COUT);
  bn_final_kernel<<<1, COUT, 0, stream>>>(partial, meanvar, COUT);
  size_t total = (size_t)M_ * COUT;
  bn_apply_kernel<<<(int)((total + 255) / 256), 256, 0, stream>>>(
      Y, meanvar, g, be, COUT);
}

extern "C" void kernel_launch(void* const* d_in, const int* in_sizes, int n_in,
                              void* d_out, int out_size, void* d_ws, size_t ws_size,
                              hipStream_t stream) {
  (void)in_sizes; (void)n_in; (void)out_size; (void)ws_size;
  const float* xyz    = (const float*)d_in[0];
  const float* points = (const float*)d_in[1];
  const float* center = (const float*)d_in[2];
  const float* tlist  = (const float*)d_in[3];
  const float *Wl[3], *bl[3], *gl[3], *bel[3];
  for (int i = 0; i < 3; ++i) {          // params flattened as (W,b,gamma,beta)x3
    Wl[i]  = (const float*)d_in[4 + 4 * i];
    bl[i]  = (const float*)d_in[5 + 4 * i];
    gl[i]  = (const float*)d_in[6 + 4 * i];
    bel[i] = (const float*)d_in[7 + 4 * i];
  }
  float* out = (float*)d_out;

  // workspace layout (~104 MB)
  char*  ws      = (char*)d_ws;
  int*   idx_ws  = (int*)ws;
  size_t off     = (size_t)2 * B_ * S_ * NS_ * sizeof(int);
  float* bufA    = (float*)(ws + off); off += (size_t)M_ * K0P_ * sizeof(float);
  float* bufB    = (float*)(ws + off); off += (size_t)M_ * 128 * sizeof(float);
  float* partial = (float*)(ws + off); off += (size_t)NCHUNK * 128 * 2 * sizeof(float);
  float* meanvar = (float*)(ws + off); off += (size_t)128 * 2 * sizeof(float);
  float* Wpad    = (float*)(ws + off);  // 128 * 68 floats max

  { // neighbor selection for both branches
    int total = 2 * B_ * S_;
    query_ball_kernel<<<(total + 255) / 256, 256, 0, stream>>>(
        xyz, center, tlist, idx_ws);
  }
  center_out_kernel<<<(2 * B_ * S_ * 3 + 255) / 256, 256, 0, stream>>>(center, out);

  for (int br = 0; br < 2; ++br) {
    const int* idx = idx_ws + (size_t)br * B_ * S_ * NS_;
    size_t gt = (size_t)M_ * K0P_;
    gather_kernel<<<(int)((gt + 255) / 256), 256, 0, stream>>>(
        xyz, points, center, idx, bufA);

    run_layer<K0_, 64>(bufA, Wl[0], bl[0], gl[0], bel[0], bufB,
                       Wpad, partial, meanvar, stream);      // 67(->68) -> 64
    run_layer<64, 64>(bufB, Wl[1], bl[1], gl[1], bel[1], bufA,
                      Wpad, partial, meanvar, stream);       // 64 -> 64
    run_layer<64, 128>(bufA, Wl[2], bl[2], gl[2], bel[2], bufB,
                       Wpad, partial, meanvar, stream);      // 64 -> 128

    maxpool_out_kernel<<<(B_ * S_ * 128 + 255) / 256, 256, 0, stream>>>(
        bufB, out, br);
  }
}